// RGNNLayer_38019050504274
// MI455X (gfx1250) — compile-verified
//
#include <hip/hip_runtime.h>
#include <math.h>

// RGNN layer for MI455X (gfx1250, wave32).
//   out = x @ rootW + rootb
//   per relation r: h = x @ Ws[r]; agg = scatter_max(h[src] -> tgt); out += finite(agg)
//
// Strategy: process one relation at a time so h (51.2 MB) stays resident in the
// 192 MB L2 for both the gather and the fmax-atomic scatter. GEMMs use
// V_WMMA_F32_16X16X4_F32 (exact f32 math; compute is negligible vs memory).
// Scatter uses native non-returning GLOBAL_ATOMIC_MAX_NUM_F32 at device scope.

typedef __attribute__((ext_vector_type(2))) float v2f;
typedef __attribute__((ext_vector_type(8))) float v8f;

#define FDIM 128
#define LDSW 132          // padded LDS row stride in floats: bank-conflict-free b64 reads
#define ROWS_PER_BLOCK 64 // 4 waves x 16 rows

// ---------------------------------------------------------------------------
// D(64x128) = X(64x128) * W(128x128) (+ bias) using v_wmma_f32_16x16x4_f32.
// W staged transposed in LDS as Wt[col][k]; X tile staged as Xs[row][k].
// ---------------------------------------------------------------------------
__global__ __launch_bounds__(128) void gemm_wmma_kernel(
    const float* __restrict__ X, const float* __restrict__ W,
    const float* __restrict__ bias, float* __restrict__ Out, int N)
{
  extern __shared__ float smem[];
  float* Wt = smem;                 // [128][LDSW]  Wt[col][k]
  float* Xs = smem + FDIM * LDSW;   // [64][LDSW]   Xs[row][k]

  const int tid  = threadIdx.x;            // 0..127
  const int wave = tid >> 5;               // 0..3
  const int lane = tid & 31;
  const int l16  = lane & 15;
  const int kh   = (lane >> 4) << 1;       // K sub-offset per lane-half: 0 or 2
  const int rowBase = blockIdx.x * ROWS_PER_BLOCK;

  // Stage W transposed: W row-major [k][col] -> Wt[col][k] (coalesced b128 reads).
  for (int i = tid; i < FDIM * (FDIM / 4); i += 128) {
    int k  = i >> 5;              // 0..127
    int c4 = (i & 31) << 2;       // 0..124 step 4
    float4 w4 = *(const float4*)(W + (size_t)k * FDIM + c4);
    Wt[(c4 + 0) * LDSW + k] = w4.x;
    Wt[(c4 + 1) * LDSW + k] = w4.y;
    Wt[(c4 + 2) * LDSW + k] = w4.z;
    Wt[(c4 + 3) * LDSW + k] = w4.w;
  }
  // Stage 64 rows of X (row clamped so out-of-range tiles read valid memory).
  for (int i = tid; i < ROWS_PER_BLOCK * (FDIM / 4); i += 128) {
    int rr  = i >> 5;             // 0..63
    int c4  = (i & 31) << 2;      // 0..124 step 4
    int row = rowBase + rr;
    row = row < N ? row : N - 1;
    *(float4*)(Xs + rr * LDSW + c4) = *(const float4*)(X + (size_t)row * FDIM + c4);
  }
  __syncthreads();

  v8f acc[8] = {};  // 8 column tiles of 16 -> full 128-wide output per wave

  const float* xrow = Xs + (wave * 16 + l16) * LDSW + kh;
#pragma unroll 4
  for (int k = 0; k < FDIM; k += 4) {
    // A fragment 16x4: lanes 0-15 hold (row, K=k..k+1), lanes 16-31 (row, K=k+2..k+3)
    v2f a = *(const v2f*)(xrow + k);
#pragma unroll
    for (int nt = 0; nt < 8; ++nt) {
      // B fragment 4x16: lanes 0-15 hold (K=k..k+1, col), lanes 16-31 (K=k+2..k+3, col)
      v2f b = *(const v2f*)(Wt + (nt * 16 + l16) * LDSW + k + kh);
      acc[nt] = __builtin_amdgcn_wmma_f32_16x16x4_f32(
          false, a, false, b, (short)0, acc[nt], false, false);
    }
  }

  // D layout: VGPR v, lanes 0-15 -> D[v][n]; lanes 16-31 -> D[v+8][n]
  const int mBase = rowBase + wave * 16 + ((lane >> 4) << 3);
#pragma unroll
  for (int nt = 0; nt < 8; ++nt) {
    const int col = nt * 16 + l16;
    const float bb = bias ? bias[col] : 0.0f;
#pragma unroll
    for (int m = 0; m < 8; ++m) {
      int row = mBase + m;
      if (row < N) Out[(size_t)row * FDIM + col] = acc[nt][m] + bb;
    }
  }
}

// ---------------------------------------------------------------------------
// Native CDNA5 float max atomic: non-returning (TH[0]=0 -> no return traffic),
// device scope so all workgroups' updates to agg are coherent.
// ---------------------------------------------------------------------------
__device__ __forceinline__ void atomic_fmax_dev(float* p, float v) {
  asm volatile("global_atomic_max_num_f32 %0, %1, off scope:SCOPE_DEV"
               :
               : "v"(p), "v"(v)
               : "memory");
}

// One wave per edge: b128 gather of h[src] (512 B/wave, L2-resident) then
// 4 device-scope float max atomics per lane into agg[tgt].
__global__ __launch_bounds__(256) void scatter_max_kernel(
    const float* __restrict__ H, const int* __restrict__ src,
    const int* __restrict__ tgt, float* __restrict__ Agg, int E)
{
  const int wave = blockIdx.x * (blockDim.x >> 5) + (threadIdx.x >> 5);
  const int lane = threadIdx.x & 31;
  if (wave >= E) return;
  const int s = src[wave];
  const int t = tgt[wave];
  const float4 v = *(const float4*)(H + (size_t)s * FDIM + lane * 4);
  float* p = Agg + (size_t)t * FDIM + lane * 4;
  atomic_fmax_dev(p + 0, v.x);
  atomic_fmax_dev(p + 1, v.y);
  atomic_fmax_dev(p + 2, v.z);
  atomic_fmax_dev(p + 3, v.w);
}

// ---------------------------------------------------------------------------
__global__ __launch_bounds__(256) void init_agg_kernel(float4* __restrict__ agg, int n4)
{
  int i = blockIdx.x * blockDim.x + threadIdx.x;
  if (i < n4) agg[i] = make_float4(-INFINITY, -INFINITY, -INFINITY, -INFINITY);
}

// out += where(isfinite(agg), agg, 0); re-arm agg to -inf for the next relation.
__global__ __launch_bounds__(256) void accum_kernel(
    float4* __restrict__ agg, float4* __restrict__ out, int n4)
{
  int i = blockIdx.x * blockDim.x + threadIdx.x;
  if (i >= n4) return;
  float4 a = agg[i];
  float4 o = out[i];
  o.x += __builtin_isfinite(a.x) ? a.x : 0.0f;
  o.y += __builtin_isfinite(a.y) ? a.y : 0.0f;
  o.z += __builtin_isfinite(a.z) ? a.z : 0.0f;
  o.w += __builtin_isfinite(a.w) ? a.w : 0.0f;
  out[i] = o;
  agg[i] = make_float4(-INFINITY, -INFINITY, -INFINITY, -INFINITY);
}

// ---------------------------------------------------------------------------
extern "C" void kernel_launch(void* const* d_in, const int* in_sizes, int n_in,
                              void* d_out, int out_size, void* d_ws, size_t ws_size,
                              hipStream_t stream)
{
  const float* x     = (const float*)d_in[0];  // [N, 128]
  const int*   edges = (const int*)  d_in[1];  // [R, 2, E]
  const float* Ws    = (const float*)d_in[2];  // [R, 128, 128]
  const float* rootW = (const float*)d_in[3];  // [128, 128]
  const float* rootb = (const float*)d_in[4];  // [128]
  float*       out   = (float*)d_out;          // [N, 128]

  const int R = 4;
  const int N = in_sizes[0] / FDIM;
  const int E = in_sizes[1] / (R * 2);

  float* h   = (float*)d_ws;                   // [N,128] per-relation transform
  float* agg = h + (size_t)N * FDIM;           // [N,128] scatter-max accumulator

  const int gemmBlocks = (N + ROWS_PER_BLOCK - 1) / ROWS_PER_BLOCK;
  const int smemBytes  = (FDIM + ROWS_PER_BLOCK) * LDSW * (int)sizeof(float);
  const int n4         = N * (FDIM / 4);
  const int vecBlocks  = (n4 + 255) / 256;
  const int edgeBlocks = (E + 7) / 8;          // 8 waves (edges) per 256-thread block

  init_agg_kernel<<<vecBlocks, 256, 0, stream>>>((float4*)agg, n4);
  gemm_wmma_kernel<<<gemmBlocks, 128, smemBytes, stream>>>(x, rootW, rootb, out, N);

  for (int r = 0; r < R; ++r) {
    const float* Wr  = Ws + (size_t)r * FDIM * FDIM;
    const int*   src = edges + (size_t)r * 2 * E;
    const int*   tgt = src + E;
    gemm_wmma_kernel<<<gemmBlocks, 128, smemBytes, stream>>>(x, Wr, nullptr, h, N);
    scatter_max_kernel<<<edgeBlocks, 256, 0, stream>>>(h, src, tgt, agg, E);
    accum_kernel<<<vecBlocks, 256, 0, stream>>>((float4*)agg, (float4*)out, n4);
  }
}